// ConsistencyFullRNNTLoss_12884901888035
// MI455X (gfx1250) — compile-verified
//
#include <hip/hip_runtime.h>
#include <hip/hip_bf16.h>
#include <stdint.h>

// Problem dims fixed by the reference setup_inputs().
#define B_   4
#define T_   256
#define U1_  65
#define V_   1025
#define NTHR 256

typedef __attribute__((ext_vector_type(2))) float        v2f;
typedef __attribute__((ext_vector_type(8))) float        v8f;
typedef __attribute__((ext_vector_type(4))) unsigned int u32x4;
typedef __attribute__((ext_vector_type(8))) int          i32x8;
typedef __attribute__((ext_vector_type(4))) int          i32x4;

// ---------------------------------------------------------------------------
// Stage one row of V_ floats from global into LDS.
//  - Preferred: Tensor Data Mover (tensor_load_to_lds, TENSORcnt).
//    This toolchain exposes the 6-arg builtin:
//      (u32x4 g0, i32x8 g1, i32x4 g2, i32x4 g3, i32x8 g4, i32 cpol)
//  - Fallback : global_load_async_to_lds_b32 (ASYNCcnt).
// LDS byte address = low 32 bits of the generic pointer (workgroup-relative).
// ---------------------------------------------------------------------------
#if __has_builtin(__builtin_amdgcn_tensor_load_to_lds)
#define USE_TDM 1
__device__ __forceinline__ void tdm_load_row(unsigned lds_addr, const float* gsrc) {
    unsigned long long ga = (unsigned long long)(uintptr_t)gsrc;
    // D# group0: count=1, lds_addr, global_addr[56:0], type=2 ("image")
    u32x4 g0;
    g0.x = 1u;                    // count = 1 valid descriptor
    g0.y = lds_addr;              // LDS byte address
    g0.z = (unsigned)(ga & 0xFFFFFFFFu);
    g0.w = (unsigned)(ga >> 32) | (2u << 30);   // addr[56:32] | type=2
    // D# group1: wg_mask=0, data_size=2 (4B), 1-D tile: tensor_dim0 = tile_dim0 = V_
    i32x8 g1;
    g1[0] = (int)(2u << 16);          // data_size = 4 bytes
    g1[1] = (int)((unsigned)V_ << 16);// tensor_dim0[15:0] in bits 63:48
    g1[2] = (int)(1u << 16);          // tensor_dim0 hi=0 | tensor_dim1[15:0]=1
    g1[3] = (int)((unsigned)V_ << 16);// tensor_dim1 hi=0 | tile_dim0 = V_
    g1[4] = 0;                        // tile_dim1 = 0 (unused), tile_dim2 = 0
    g1[5] = (int)V_;                  // tensor_dim0_stride[31:0]
    g1[6] = 0;
    g1[7] = 0;
    i32x4 gz4 = (i32x4)0;             // groups 2/3 unused (1-D tile)
    i32x8 gz8 = (i32x8)0;             // extra group in 6-arg builtin
    __builtin_amdgcn_tensor_load_to_lds(g0, g1, gz4, gz4, gz8, 0);
}
#else
#define USE_TDM 0
#endif

__device__ __forceinline__ void wait_tensorcnt0() {
#if __has_builtin(__builtin_amdgcn_s_wait_tensorcnt)
    __builtin_amdgcn_s_wait_tensorcnt(0);
#else
    asm volatile("s_wait_tensorcnt 0x0" ::: "memory");
#endif
}
__device__ __forceinline__ void wait_asynccnt0() {
#if __has_builtin(__builtin_amdgcn_s_wait_asynccnt)
    __builtin_amdgcn_s_wait_asynccnt(0);
#else
    asm volatile("s_wait_asynccnt 0x0" ::: "memory");
#endif
}

// ---------------------------------------------------------------------------
// One block per lattice cell (b,t,u). Early-exit on the RNNT mask (~74% of
// rows are masked out -> skip their 8.2 KB entirely). Valid rows: stage both
// logit rows into LDS once, run two passes (max, then exp-weighted sums) from
// LDS, reduce with wave32 shuffles + one v_wmma_f32_16x16x4_f32, and
// atomically accumulate the per-batch KL sum.
// ---------------------------------------------------------------------------
__global__ __launch_bounds__(NTHR)
void kl_rows_kernel(const float* __restrict__ t_logits,
                    const float* __restrict__ s_logits,
                    const int*   __restrict__ src_len,
                    const int*   __restrict__ tgt_len,
                    float*       __restrict__ acc) {
    const int row = blockIdx.x;
    const int b   = row / (T_ * U1_);
    const int rem = row - b * (T_ * U1_);
    const int t   = rem / U1_;
    const int u   = rem - t * U1_;
    if (t >= src_len[b] || u > tgt_len[b]) return;   // masked cell: zero traffic

    __shared__ float sT[V_ + 3];
    __shared__ float sS[V_ + 3];
    __shared__ float wmax[16];      // per-wave maxima: [0..7]=teacher, [8..15]=student
    __shared__ float wred[8 * 4];   // per-wave partial sums: wave*4 + {Zt,Zs,Bt,Bs}

    const int    tid  = threadIdx.x;
    const int    wave = tid >> 5;
    const size_t base = (size_t)row * V_;
    const unsigned ldsT = (unsigned)(uintptr_t)(void*)sT;
    const unsigned ldsS = (unsigned)(uintptr_t)(void*)sS;

#if USE_TDM
    if (tid < 32) {                         // one wave issues the two DMAs
        tdm_load_row(ldsT, t_logits + base);
        tdm_load_row(ldsS, s_logits + base);
        wait_tensorcnt0();
    }
    __syncthreads();
#else
    for (int i = tid; i < V_; i += NTHR) {
        const float* gt = t_logits + base + i;
        const float* gs = s_logits + base + i;
        unsigned lt = ldsT + 4u * (unsigned)i;
        unsigned ls = ldsS + 4u * (unsigned)i;
        asm volatile("global_load_async_to_lds_b32 %0, %1, off"
                     :: "v"(lt), "v"(gt) : "memory");
        asm volatile("global_load_async_to_lds_b32 %0, %1, off"
                     :: "v"(ls), "v"(gs) : "memory");
    }
    wait_asynccnt0();
    __syncthreads();
#endif

    // ---- pass 1: row maxima (from LDS) ----
    float mt = -3.402823466e38f, ms = -3.402823466e38f;
    for (int i = tid; i < V_; i += NTHR) {
        mt = fmaxf(mt, sT[i]);
        ms = fmaxf(ms, sS[i]);
    }
    for (int o = 16; o; o >>= 1) {
        mt = fmaxf(mt, __shfl_xor(mt, o, 32));
        ms = fmaxf(ms, __shfl_xor(ms, o, 32));
    }
    if ((tid & 31) == 0) { wmax[wave] = mt; wmax[8 + wave] = ms; }
    __syncthreads();
    mt = wmax[0]; ms = wmax[8];
    #pragma unroll
    for (int w = 1; w < 8; ++w) {
        mt = fmaxf(mt, wmax[w]);
        ms = fmaxf(ms, wmax[8 + w]);
    }

    // ---- pass 2: Zt=Σe^dt, Zs=Σe^ds, Bt=Σe^dt·g, Bs=-Σe^ds·g, g=dt-ds ----
    // (symmetric KL: log-partition terms cancel, kl = .5*(Bt/Zt + Bs/Zs))
    float zt = 0.f, zs = 0.f, bt = 0.f, bs = 0.f;
    for (int i = tid; i < V_; i += NTHR) {
        float dt = sT[i] - mt;
        float ds = sS[i] - ms;
        float g  = dt - ds;
        float et = __expf(dt);
        float es = __expf(ds);
        zt += et; zs += es;
        bt = __fmaf_rn(et,  g, bt);
        bs = __fmaf_rn(es, -g, bs);
    }
    for (int o = 16; o; o >>= 1) {
        zt += __shfl_xor(zt, o, 32);
        zs += __shfl_xor(zs, o, 32);
        bt += __shfl_xor(bt, o, 32);
        bs += __shfl_xor(bs, o, 32);
    }
    if ((tid & 31) == 0) {
        wred[wave * 4 + 0] = zt; wred[wave * 4 + 1] = zs;
        wred[wave * 4 + 2] = bt; wred[wave * 4 + 3] = bs;
    }
    __syncthreads();

    // ---- cross-wave reduce of 8x4 partials via one WMMA (B = ones) ----
    // A(16x4) layout: lane m holds A[m,0],A[m,1]; lane m+16 holds A[m,2],A[m,3].
    // Set A[m,k] = wred[k][m] + wred[k+4][m]  (m<4) so D[m,n] = Σ_k A[m,k] = total_m.
    if (tid < 32) {
        const int m  = tid & 15;
        const int k0 = (tid < 16) ? 0 : 2;
        float a0 = 0.f, a1 = 0.f;
        if (m < 4) {
            a0 = wred[k0 * 4 + m]       + wred[(k0 + 4) * 4 + m];
            a1 = wred[(k0 + 1) * 4 + m] + wred[(k0 + 5) * 4 + m];
        }
        v2f A; A.x = a0; A.y = a1;
        v2f Bm; Bm.x = 1.0f; Bm.y = 1.0f;
        v8f C = {};
        C = __builtin_amdgcn_wmma_f32_16x16x4_f32(
                /*neg_a=*/false, A, /*neg_b=*/false, Bm,
                /*c_mod=*/(short)0, C, /*reuse_a=*/false, /*reuse_b=*/false);
        if (tid == 0) {
            // C/D layout: VGPR r, lanes 0-15 = D[r][n]  ->  lane0 C[q] = total_q
            float Zt = C[0], Zs = C[1], Bt = C[2], Bs = C[3];
            float kl = 0.5f * (Bt / Zt + Bs / Zs);
            atomicAdd(&acc[b], kl);
        }
    }
}

__global__ void zero_acc_kernel(float* __restrict__ acc) {
    if (threadIdx.x < B_) acc[threadIdx.x] = 0.0f;
}

__global__ void finalize_kernel(const float* __restrict__ acc,
                                const int*   __restrict__ src_len,
                                const int*   __restrict__ tgt_len,
                                float*       __restrict__ out) {
    if (threadIdx.x == 0) {
        float s = 0.0f;
        #pragma unroll
        for (int b = 0; b < B_; ++b) {
            float denom = fmaxf(1.0f, (float)src_len[b] * (float)(tgt_len[b] + 1));
            s += acc[b] / denom;
        }
        out[0] = s / (float)B_;
    }
}

extern "C" void kernel_launch(void* const* d_in, const int* in_sizes, int n_in,
                              void* d_out, int out_size, void* d_ws, size_t ws_size,
                              hipStream_t stream) {
    const float* t_logits = (const float*)d_in[0];
    const float* s_logits = (const float*)d_in[1];
    const int*   src_len  = (const int*)d_in[2];
    const int*   tgt_len  = (const int*)d_in[3];
    float*       out      = (float*)d_out;
    float*       acc      = (float*)d_ws;   // B_ floats of scratch

    zero_acc_kernel<<<1, 64, 0, stream>>>(acc);
    kl_rows_kernel<<<B_ * T_ * U1_, NTHR, 0, stream>>>(t_logits, s_logits,
                                                       src_len, tgt_len, acc);
    finalize_kernel<<<1, 32, 0, stream>>>(acc, src_len, tgt_len, out);
}